// BoundaryGreenBranch_25735444038337
// MI455X (gfx1250) — compile-verified
//
#include <hip/hip_runtime.h>

typedef float v2f __attribute__((ext_vector_type(2)));
typedef float v8f __attribute__((ext_vector_type(8)));

#define B_      4
#define NBC     128
#define HID     64
#define NINT    4096
#define CHUNK   32            // boundary points per chunk
#define NCHUNK  4             // NBC / CHUNK
#define NSPLIT  2             // column halves per j-tile (one wave each)
#define JPB     64            // interior points per block (8 waves = 4 j-tiles x 2 col-halves)
#define JBLOCKS (NINT / JPB)  // 64

// workspace layout (float offsets)
#define WS_APRIME 0
#define WS_CFEAT  (B_ * NBC * HID)                 // 32768
#define WS_PART   (WS_CFEAT + B_ * NINT * HID)     // 1081344
#define NSLICE    (NCHUNK * NSPLIT)                // 8 partial slices
// total floats: 1081344 + 8*B_*NINT = 1212416 (~4.85 MB)

// ---------------------------------------------------------------------------
// Kernel 1a: boundary encoder -> a' = relu(relu(x@W0+b0)@W1+b1)@G0w[:64] + G0b
// ---------------------------------------------------------------------------
__global__ void boundary_encode(const float* __restrict__ binfo,
                                const float* __restrict__ W0, const float* __restrict__ b0,
                                const float* __restrict__ W1, const float* __restrict__ b1,
                                const float* __restrict__ G0w, const float* __restrict__ G0b,
                                float* __restrict__ aprime)
{
    __shared__ float sh[HID];
    __shared__ float sin3[3];
    const int row = blockIdx.x;        // b*NBC + i
    const int n   = threadIdx.x;       // 0..63
    if (n < 3) sin3[n] = binfo[row * 3 + n];
    __syncthreads();
    float x = b0[n];
    for (int m = 0; m < 3; ++m) x = fmaf(sin3[m], W0[m * HID + n], x);
    x = fmaxf(x, 0.f);
    sh[n] = x;
    __syncthreads();
    float y = b1[n];
    for (int k = 0; k < HID; ++k) y = fmaf(sh[k], W1[k * HID + n], y);
    y = fmaxf(y, 0.f);
    __syncthreads();
    sh[n] = y;
    __syncthreads();
    float z = G0b[n];                  // fold G0b into a'
    for (int k = 0; k < HID; ++k) z = fmaf(sh[k], G0w[k * HID + n], z);
    aprime[row * HID + n] = z;
}

// ---------------------------------------------------------------------------
// Kernel 1b: c = interior_coords @ G0w[64:66]   (B, NINT, HID)
// ---------------------------------------------------------------------------
__global__ void coord_feat(const float* __restrict__ ic,
                           const float* __restrict__ G0w,
                           float* __restrict__ cfeat)
{
    const int idx = blockIdx.x * blockDim.x + threadIdx.x;  // B_*NINT*HID threads
    const int n = idx & (HID - 1);
    const int r = idx >> 6;            // b*NINT + j
    const float x = ic[r * 2 + 0];
    const float y = ic[r * 2 + 1];
    cfeat[idx] = x * G0w[64 * HID + n] + y * G0w[65 * HID + n];
}

// ---------------------------------------------------------------------------
// Kernel 2: fused Green-kernel core.
// Wave pair (nhalf=0/1) shares one 16-row j-tile; each wave owns 2 of the 4
// 16-column tiles of G1w (64 Breg VGPRs) so total pressure stays < 256.
// Per boundary point i:
//   A_t = relu(c_tile + a'_i)                       (regs, WMMA A layout)
//   acc[n] = A_0 x B[n][0] + Cinit[n]  (Cinit = persistent G1b broadcast;
//            no per-iteration accumulator init moves)
//   acc[n] += A_t x B[n][t]   for t=1..15           (v_wmma_f32_16x16x4_f32)
//   rowacc  += relu(acc) * g2                       (per-lane FMA)
// Cross-lane butterfly -> partial[chunk*2+nhalf][b][j].
//
// Layout assumptions (CDNA5 ISA 7.12.2, 32-bit operands, wave32):
//   A 16x4 : lane<16 -> (M=lane, K={4t,4t+1}), lane>=16 -> (M=lane-16, K={4t+2,4t+3})
//   B 4x16 : VGPR0/1 hold K pair {k0,k0+1}; k0 = 4t + 2*(lane>=16); lane&15 = N
//   C/D    : VGPR v: lane<16 -> (M=v, N=lane), lane>=16 -> (M=v+8, N=lane-16)
// ---------------------------------------------------------------------------
__global__ __launch_bounds__(256)
void green_main(const float* __restrict__ aprime,
                const float* __restrict__ cfeat,
                const float* __restrict__ G1w,
                const float* __restrict__ G1b,
                const float* __restrict__ G2w,
                float* __restrict__ partial)
{
    __shared__ float lds_a[CHUNK * HID];   // 8 KB: a' rows for this chunk

    const int tid   = threadIdx.x;
    const int lane  = tid & 31;
    const int wave  = tid >> 5;            // 0..7
    const int li    = lane & 15;
    const int half  = lane >> 4;
    const int jt    = wave >> 1;           // 0..3 : j-tile within block
    const int nhalf = wave & 1;            // 0..1 : which pair of column tiles

    const int blk    = blockIdx.x;
    const int jblock = blk % JBLOCKS;
    const int b      = (blk / JBLOCKS) % B_;
    const int chunk  = blk / (JBLOCKS * B_);

    // stage a' chunk into LDS cooperatively (shared by all 8 waves)
    {
        const float* src = aprime + (size_t)(b * NBC + chunk * CHUNK) * HID;
        for (int idx = tid; idx < CHUNK * HID; idx += 256)
            lds_a[idx] = src[idx];
    }
    __syncthreads();

    const int jbase = jblock * JPB + jt * 16;
    const int jrow  = jbase + li;

    // c tile in A layout (reused across all 32 boundary points)
    v2f cA[16];
    {
        const float* crow = cfeat + (size_t)(b * NINT + jrow) * HID;
#pragma unroll
        for (int t = 0; t < 16; ++t) {
            const int k0 = 4 * t + 2 * half;
            cA[t] = *(const v2f*)(crow + k0);
        }
    }

    // this wave's 2 column tiles of G1w, resident in registers in B layout
    v2f Breg[NSPLIT][16];
#pragma unroll
    for (int nn = 0; nn < NSPLIT; ++nn) {
        const int col = (nhalf * NSPLIT + nn) * 16 + li;
#pragma unroll
        for (int t = 0; t < 16; ++t) {
            const int k0 = 4 * t + 2 * half;
            v2f v;
            v.x = G1w[k0 * HID + col];
            v.y = G1w[(k0 + 1) * HID + col];
            Breg[nn][t] = v;
        }
    }

    // persistent C-init (G1b broadcast) and g2 broadcast; column N = tile*16 + li
    v8f Cinit[NSPLIT];
    float g2v[NSPLIT];
#pragma unroll
    for (int nn = 0; nn < NSPLIT; ++nn) {
        const int colbase = (nhalf * NSPLIT + nn) * 16 + li;
        const float gb = G1b[colbase];
        g2v[nn] = G2w[colbase];
        v8f t0;
#pragma unroll
        for (int v = 0; v < 8; ++v) t0[v] = gb;
        Cinit[nn] = t0;
    }

    float rowacc[8];
#pragma unroll
    for (int v = 0; v < 8; ++v) rowacc[v] = 0.f;

    for (int ii = 0; ii < CHUNK; ++ii) {
        const float* arow = &lds_a[ii * HID];
        v8f acc[NSPLIT];

        // t = 0: seed accumulators straight from Cinit (no mov churn)
        {
            const int k0 = 2 * half;
            const v2f a2 = *(const v2f*)(arow + k0);
            v2f h;
            h.x = fmaxf(cA[0].x + a2.x, 0.f);
            h.y = fmaxf(cA[0].y + a2.y, 0.f);
#pragma unroll
            for (int nn = 0; nn < NSPLIT; ++nn)
                acc[nn] = __builtin_amdgcn_wmma_f32_16x16x4_f32(
                    false, h, false, Breg[nn][0], (short)0, Cinit[nn], false, false);
        }

#pragma unroll
        for (int t = 1; t < 16; ++t) {
            const int k0 = 4 * t + 2 * half;
            const v2f a2 = *(const v2f*)(arow + k0);   // ds_load_b64 broadcast
            v2f h;
            h.x = fmaxf(cA[t].x + a2.x, 0.f);          // h1 = relu(a + c + G0b)
            h.y = fmaxf(cA[t].y + a2.y, 0.f);
#pragma unroll
            for (int nn = 0; nn < NSPLIT; ++nn)
                acc[nn] = __builtin_amdgcn_wmma_f32_16x16x4_f32(
                    false, h, false, Breg[nn][t], (short)0, acc[nn], false, false);
        }

        // h2 = relu(acc); contribution dot with g2, accumulated per (row, lane-col)
#pragma unroll
        for (int nn = 0; nn < NSPLIT; ++nn) {
#pragma unroll
            for (int v = 0; v < 8; ++v)
                rowacc[v] = fmaf(fmaxf(acc[nn][v], 0.f), g2v[nn], rowacc[v]);
        }
    }

    // reduce each row's 16 per-lane column partials (halves are independent rows)
#pragma unroll
    for (int v = 0; v < 8; ++v) {
        float s = rowacc[v];
        s += __shfl_xor(s, 8, 32);
        s += __shfl_xor(s, 4, 32);
        s += __shfl_xor(s, 2, 32);
        s += __shfl_xor(s, 1, 32);
        if (li == 0) {
            const int j = jbase + v + 8 * half;
            partial[((size_t)(chunk * NSPLIT + nhalf) * B_ + b) * NINT + j] = s;
        }
    }
}

// ---------------------------------------------------------------------------
// Kernel 3: out[b,j] = (sum of 8 partial slices)/NBC + G2b
// ---------------------------------------------------------------------------
__global__ void green_finalize(const float* __restrict__ partial,
                               const float* __restrict__ G2b,
                               float* __restrict__ out)
{
    const int idx = blockIdx.x * blockDim.x + threadIdx.x;  // b*NINT + j
    float s = 0.f;
#pragma unroll
    for (int sl = 0; sl < NSLICE; ++sl)
        s += partial[(size_t)sl * B_ * NINT + idx];
    out[idx] = s * (1.0f / (float)NBC) + G2b[0];
}

// ---------------------------------------------------------------------------
extern "C" void kernel_launch(void* const* d_in, const int* in_sizes, int n_in,
                              void* d_out, int out_size, void* d_ws, size_t ws_size,
                              hipStream_t stream)
{
    const float* binfo = (const float*)d_in[0];   // (B, NBC, 3)
    const float* ic    = (const float*)d_in[1];   // (B, NINT, 2)
    const float* W0    = (const float*)d_in[2];   // (3, 64)
    const float* b0    = (const float*)d_in[3];
    const float* W1    = (const float*)d_in[4];   // (64, 64)
    const float* b1    = (const float*)d_in[5];
    const float* G0w   = (const float*)d_in[6];   // (66, 64)
    const float* G0b   = (const float*)d_in[7];
    const float* G1w   = (const float*)d_in[8];   // (64, 64)
    const float* G1b   = (const float*)d_in[9];
    const float* G2w   = (const float*)d_in[10];  // (64, 1)
    const float* G2b   = (const float*)d_in[11];
    // d_in[12], d_in[13]: interior_h/w (compile-time constants here)

    float* ws      = (float*)d_ws;
    float* aprime  = ws + WS_APRIME;
    float* cfeat   = ws + WS_CFEAT;
    float* partial = ws + WS_PART;
    float* out     = (float*)d_out;

    boundary_encode<<<B_ * NBC, HID, 0, stream>>>(binfo, W0, b0, W1, b1, G0w, G0b, aprime);

    coord_feat<<<(B_ * NINT * HID) / 256, 256, 0, stream>>>(ic, G0w, cfeat);

    green_main<<<NCHUNK * B_ * JBLOCKS, 256, 0, stream>>>(aprime, cfeat, G1w, G1b, G2w, partial);

    green_finalize<<<(B_ * NINT) / 256, 256, 0, stream>>>(partial, G2b, out);
}